// StableHybridGNN_23587960390232
// MI455X (gfx1250) — compile-verified
//
#include <hip/hip_runtime.h>

typedef __attribute__((ext_vector_type(2))) float v2f;
typedef __attribute__((ext_vector_type(8))) float v8f;

#define GN   50000
#define GE   800000
#define INC  128
#define CDIM 16
#define K0   (INC + CDIM)   // 144
#define HID  128
#define OUTC 64

// ---------------- degree / norm ----------------
__global__ void deg_init_kernel(float* deg) {
  int i = blockIdx.x * blockDim.x + threadIdx.x;
  if (i < GN) deg[i] = 1.0f;
}

__global__ void deg_count_kernel(const int* __restrict__ dst, float* deg) {
  int e = blockIdx.x * blockDim.x + threadIdx.x;
  if (e < GE) atomicAdd(&deg[dst[e]], 1.0f);
}

__global__ void dis_kernel(float* deg) {
  int i = blockIdx.x * blockDim.x + threadIdx.x;
  if (i < GN) deg[i] = rsqrtf(deg[i]);
}

// ---------------- utility ----------------
__global__ void zero_kernel(float* p, int n) {
  int i = blockIdx.x * blockDim.x + threadIdx.x;
  if (i < n) p[i] = 0.0f;
}

// ---------------- WMMA micro-tile: fp32 16x16x4 ----------------
// A: LDS, 16 x K row-major (ld = K). B: global, K x ldb row-major.
// Per §7.12.2: A lanes 0-15 hold K={0,1}, lanes 16-31 hold K={2,3};
// C/D: VGPR r holds M=r (lanes 0-15) and M=r+8 (lanes 16-31), N=lane&15.
__device__ __forceinline__ v8f wmma_f32_tile(const float* __restrict__ As, int K,
                                             const float* __restrict__ B, int ldb,
                                             int colbase, int lane) {
  const int half = lane >> 4;
  const int l15  = lane & 15;
  const float* arow = As + l15 * K + half * 2;
  const float* bptr = B + (half * 2) * ldb + colbase + l15;
  v8f c = {};
#pragma unroll
  for (int k = 0; k < 160; k += 4) {   // bounded by max K (unrolled; guarded)
    if (k >= K) break;
    v2f a = *(const v2f*)(arow + k);
    v2f b;
    b.x = bptr[(k + 0) * ldb];
    b.y = bptr[(k + 1) * ldb];
    c = __builtin_amdgcn_wmma_f32_16x16x4_f32(false, a, false, b, (short)0, c,
                                              false, false);
  }
  return c;
}

// ---------------- GEMM 0: concat(x, emb[comm]) @ W0 -> hp ----------------
__global__ __launch_bounds__(256) void gemm0_kernel(const float* __restrict__ x,
                                                    const int* __restrict__ comm,
                                                    const float* __restrict__ emb,
                                                    const float* __restrict__ W,
                                                    float* __restrict__ hp) {
  __shared__ float As[16 * K0];
  const int tid  = threadIdx.x;
  const int row0 = blockIdx.x * 16;
  for (int idx = tid; idx < 16 * K0; idx += 256) {
    int r = idx / K0, c = idx % K0;
    int grow = row0 + r;
    As[idx] = (c < INC) ? x[(size_t)grow * INC + c]
                        : emb[(size_t)comm[grow] * CDIM + (c - INC)];
  }
  __syncthreads();
  const int wave = tid >> 5, lane = tid & 31;
  const int half = lane >> 4, l15 = lane & 15;
  const int colbase = wave * 16;
  v8f c = wmma_f32_tile(As, K0, W, HID, colbase, lane);
#pragma unroll
  for (int r = 0; r < 8; ++r)
    hp[(size_t)(row0 + r + half * 8) * HID + colbase + l15] = c[r];
}

// ---------------- GEMM mid: hin @ W -> hp ----------------
__global__ __launch_bounds__(256) void gemm_mid_kernel(const float* __restrict__ hin,
                                                       const float* __restrict__ W,
                                                       float* __restrict__ hp) {
  __shared__ float As[16 * HID];
  const int tid  = threadIdx.x;
  const int row0 = blockIdx.x * 16;
  for (int idx = tid; idx < 16 * HID; idx += 256)
    As[idx] = hin[(size_t)row0 * HID + idx];
  __syncthreads();
  const int wave = tid >> 5, lane = tid & 31;
  const int half = lane >> 4, l15 = lane & 15;
  const int colbase = wave * 16;
  v8f c = wmma_f32_tile(As, HID, W, HID, colbase, lane);
#pragma unroll
  for (int r = 0; r < 8; ++r)
    hp[(size_t)(row0 + r + half * 8) * HID + colbase + l15] = c[r];
}

// ---------------- final GEMM: max(h1,h2,h3) @ Wp + bp -> out ----------------
__global__ __launch_bounds__(128) void gemm_jk_kernel(const float* __restrict__ h1,
                                                      const float* __restrict__ h2,
                                                      const float* __restrict__ h3,
                                                      const float* __restrict__ Wp,
                                                      const float* __restrict__ bp,
                                                      float* __restrict__ out) {
  __shared__ float As[16 * HID];
  const int tid  = threadIdx.x;
  const int row0 = blockIdx.x * 16;
  for (int idx = tid; idx < 16 * HID; idx += 128) {
    size_t g = (size_t)row0 * HID + idx;
    As[idx] = fmaxf(fmaxf(h1[g], h2[g]), h3[g]);
  }
  __syncthreads();
  const int wave = tid >> 5, lane = tid & 31;   // 4 waves cover 64 cols
  const int half = lane >> 4, l15 = lane & 15;
  const int colbase = wave * 16;
  v8f c = wmma_f32_tile(As, HID, Wp, OUTC, colbase, lane);
#pragma unroll
  for (int r = 0; r < 8; ++r)
    out[(size_t)(row0 + r + half * 8) * OUTC + colbase + l15] =
        c[r] + bp[colbase + l15];
}

// ---------------- edge scatter: acc[dst] += hp[src] * norm ----------------
__global__ void scatter_kernel(const float* __restrict__ hp,
                               const int* __restrict__ src,
                               const int* __restrict__ dst,
                               const float* __restrict__ dis,
                               float* __restrict__ acc) {
  int tid = blockIdx.x * blockDim.x + threadIdx.x;   // GE*32 threads
  int e = tid >> 5;
  if (e >= GE) return;
  int c = (tid & 31) << 2;
  int s = src[e], d = dst[e];
  float nrm = dis[s] * dis[d];
  const float4 v = *(const float4*)(hp + (size_t)s * HID + c);
  float* a = acc + (size_t)d * HID + c;
  atomicAdd(a + 0, v.x * nrm);
  atomicAdd(a + 1, v.y * nrm);
  atomicAdd(a + 2, v.z * nrm);
  atomicAdd(a + 3, v.w * nrm);
}

// ---------------- finalize: h = relu(acc + hp*dis^2 + b) ----------------
__global__ void finalize_kernel(float* __restrict__ acc,
                                const float* __restrict__ hp,
                                const float* __restrict__ dis,
                                const float* __restrict__ b) {
  int idx = blockIdx.x * blockDim.x + threadIdx.x;   // GN*HID threads
  int i = idx >> 7, f = idx & 127;
  if (i >= GN) return;
  float ds = dis[i];
  float v  = acc[idx] + hp[idx] * ds * ds + b[f];
  acc[idx] = fmaxf(v, 0.0f);
}

extern "C" void kernel_launch(void* const* d_in, const int* in_sizes, int n_in,
                              void* d_out, int out_size, void* d_ws, size_t ws_size,
                              hipStream_t stream) {
  const float* x    = (const float*)d_in[0];
  const int*   eidx = (const int*)d_in[1];   // [2,E]: row0=src, row1=dst
  const int*   comm = (const int*)d_in[2];
  const float* emb  = (const float*)d_in[3];
  const float* W0   = (const float*)d_in[4];
  const float* b0   = (const float*)d_in[5];
  const float* W1   = (const float*)d_in[6];
  const float* b1   = (const float*)d_in[7];
  const float* W2   = (const float*)d_in[8];
  const float* b2   = (const float*)d_in[9];
  const float* Wp   = (const float*)d_in[10];
  const float* bp   = (const float*)d_in[11];
  float* out = (float*)d_out;

  const int* src = eidx;
  const int* dst = eidx + GE;

  char* ws = (char*)d_ws;
  float* dis = (float*)ws;
  size_t off = (((size_t)GN * 4) + 255) & ~(size_t)255;
  const size_t FB = (size_t)GN * HID * 4;   // 25.6 MB, 256B-aligned
  float* hp = (float*)(ws + off); off += FB;
  float* h1 = (float*)(ws + off); off += FB;
  float* h2 = (float*)(ws + off); off += FB;
  float* h3 = (float*)(ws + off); off += FB;

  const int FEAT  = GN * HID;               // 6.4M
  const int gFeat = (FEAT + 255) / 256;     // 25000
  const int gRows = GN / 16;                // 3125
  const int gScat = (GE * 32) / 256;        // 100000

  // degrees -> deg_inv_sqrt
  deg_init_kernel<<<(GN + 255) / 256, 256, 0, stream>>>(dis);
  deg_count_kernel<<<(GE + 255) / 256, 256, 0, stream>>>(dst, dis);
  dis_kernel<<<(GN + 255) / 256, 256, 0, stream>>>(dis);

  // layer 1
  gemm0_kernel<<<gRows, 256, 0, stream>>>(x, comm, emb, W0, hp);
  zero_kernel<<<gFeat, 256, 0, stream>>>(h1, FEAT);
  scatter_kernel<<<gScat, 256, 0, stream>>>(hp, src, dst, dis, h1);
  finalize_kernel<<<gFeat, 256, 0, stream>>>(h1, hp, dis, b0);

  // layer 2
  gemm_mid_kernel<<<gRows, 256, 0, stream>>>(h1, W1, hp);
  zero_kernel<<<gFeat, 256, 0, stream>>>(h2, FEAT);
  scatter_kernel<<<gScat, 256, 0, stream>>>(hp, src, dst, dis, h2);
  finalize_kernel<<<gFeat, 256, 0, stream>>>(h2, hp, dis, b1);

  // layer 3
  gemm_mid_kernel<<<gRows, 256, 0, stream>>>(h2, W2, hp);
  zero_kernel<<<gFeat, 256, 0, stream>>>(h3, FEAT);
  scatter_kernel<<<gScat, 256, 0, stream>>>(hp, src, dst, dis, h3);
  finalize_kernel<<<gFeat, 256, 0, stream>>>(h3, hp, dis, b2);

  // JK max + projection
  gemm_jk_kernel<<<gRows, 128, 0, stream>>>(h1, h2, h3, Wp, bp, out);
}